// VariableEmbedding_8890582303479
// MI455X (gfx1250) — compile-verified
//
#include <hip/hip_runtime.h>

// y[b,t,v,d] = x[b,t,v] * W[v,d] + b[v,d]
// B=64, T=288, V=24, D=256  ->  453 MB f32 out, store-bandwidth bound (~19.5us @ 23.3 TB/s).
// Implemented as rank-1 outer-product + bias via V_WMMA_F32_16X16X4_F32 per 16x16 tile.

typedef __attribute__((ext_vector_type(2))) float v2f;
typedef __attribute__((ext_vector_type(8))) float v8f;

#define B_SZ 64
#define T_SZ 288
#define V_SZ 24
#define D_SZ 256
#define BT_SZ (B_SZ * T_SZ)          // 18432 (divisible by 16)
#define ROW_STRIDE (V_SZ * D_SZ)     // 6144 floats between consecutive bt rows
#define WAVES_PER_BLOCK 8

__global__ __launch_bounds__(256) void varembed_wmma_f32(
    const float* __restrict__ x,     // [BT, V]
    const float* __restrict__ W,     // [V, D]
    const float* __restrict__ bias,  // [V, D]
    float* __restrict__ out)         // [BT, V, D]
{
    const int lane      = threadIdx.x & 31;        // wave32
    const int waveInBlk = threadIdx.x >> 5;
    const int gid       = blockIdx.x * WAVES_PER_BLOCK + waveInBlk;

    const int btTile = gid / V_SZ;                 // 0 .. 1151
    const int v      = gid - btTile * V_SZ;        // 0 .. 23
    const int bt0    = btTile * 16;

    const int  n  = lane & 15;                     // column within tile / row index for A load
    const bool lo = (lane < 16);

    // ---- A matrix 16x4 (MxK), only K=0 non-zero: A[m][0] = x[bt0+m, v] ----
    // 32-bit A layout: lanes 0-15 hold {K=0,K=1} in VGPR{0,1}; lanes 16-31 hold {K=2,K=3}.
    const float xv = x[(size_t)(bt0 + n) * V_SZ + v];   // uniform-ish gather, x is tiny/L2-hot
    v2f A;
    A.x = lo ? xv : 0.0f;   // K=0 (lanes 0-15) ; K=2 (lanes 16-31) forced 0
    A.y = 0.0f;             // K=1 / K=3 = 0

    const float* Wrow = W    + (size_t)v * D_SZ;
    const float* brow = bias + (size_t)v * D_SZ;

    // Store base: D-matrix layout -> lane holds col n, rows m = r + 8*(lane>=16) in VGPR r.
    float* p0 = out + (size_t)(bt0 + (lo ? 0 : 8)) * ROW_STRIDE
                    + (size_t)v * D_SZ + n;

    for (int d0 = 0; d0 < D_SZ; d0 += 16) {
        const float wv = Wrow[d0 + n];
        const float bv = brow[d0 + n];

        // ---- B matrix 4x16 (KxN), only row K=0 non-zero: B[0][n] = W[v, d0+n] ----
        v2f Bm;
        Bm.x = lo ? wv : 0.0f;
        Bm.y = 0.0f;

        // ---- C matrix: bias broadcast across all 16 rows (all 8 acc VGPRs) ----
        v8f C;
        #pragma unroll
        for (int i = 0; i < 8; ++i) C[i] = bv;

        // D = A x B + C  (exact f32, RNE)
        v8f D = __builtin_amdgcn_wmma_f32_16x16x4_f32(
            /*neg_a=*/false, A, /*neg_b=*/false, Bm,
            /*c_mod=*/(short)0, C, /*reuse_a=*/false, /*reuse_b=*/false);

        // 8 stores, each 32 lanes x 4B = 128B (two 64B contiguous runs);
        // row stride folds into the 24-bit instruction offset (r * 24576 B).
        float* p = p0 + d0;
        #pragma unroll
        for (int r = 0; r < 8; ++r)
            p[(size_t)r * ROW_STRIDE] = D[r];
    }
}

extern "C" void kernel_launch(void* const* d_in, const int* in_sizes, int n_in,
                              void* d_out, int out_size, void* d_ws, size_t ws_size,
                              hipStream_t stream) {
    const float* x  = (const float*)d_in[0];   // (B,T,V) f32
    const float* W  = (const float*)d_in[1];   // (V,D)   f32
    const float* b  = (const float*)d_in[2];   // (V,D)   f32
    float*       out = (float*)d_out;          // (B,T,V,D) f32

    const int totalWaves = (BT_SZ / 16) * V_SZ;          // 1152 * 24 = 27648
    const int blocks     = totalWaves / WAVES_PER_BLOCK; // 3456

    hipLaunchKernelGGL(varembed_wmma_f32, dim3(blocks), dim3(256), 0, stream,
                       x, W, b, out);
}